// OutputPPBlock_54941221650645
// MI455X (gfx1250) — compile-verified
//
#include <hip/hip_runtime.h>

typedef __attribute__((ext_vector_type(2))) float v2f;
typedef __attribute__((ext_vector_type(8))) float v8f;

#define HIDDEN 128
#define NRAD   6
#define NTILES (HIDDEN / 16)   // 8 N-tiles of 16 hidden channels

// ---------------------------------------------------------------------------
// Zero the output (harness poisons d_out with 0xAA). float4 stores; also
// warms the 51 MB output into L2 ahead of the atomic storm.
// ---------------------------------------------------------------------------
__global__ void zero_out_kernel(float* __restrict__ out, int n) {
  const int n4     = n >> 2;
  const int stride = gridDim.x * blockDim.x;
  float4* __restrict__ out4 = (float4*)out;
  for (int i = blockIdx.x * blockDim.x + threadIdx.x; i < n4; i += stride)
    out4[i] = make_float4(0.f, 0.f, 0.f, 0.f);
  // tail (n % 4)
  for (int i = (n4 << 2) + blockIdx.x * blockDim.x + threadIdx.x; i < n;
       i += stride)
    out[i] = 0.0f;
}

// ---------------------------------------------------------------------------
// Main kernel: per wave, a 16-edge x 128-hidden tile.
//   einsum [16x6]x[6x128] via V_WMMA_F32_16X16X4_F32 (K padded 6->8, 2 steps),
//   then elementwise * x (non-temporal load) and global_atomic_add_f32
//   scatter (non-returning, device scope).
//
// WMMA f32 16x16x4 register layouts (CDNA5 ISA):
//   A (16x4, MxK): lane = M (lanes 16-31 repeat M=0..15);
//                  VGPR0 = K0 (lanes 0-15) / K2 (lanes 16-31); VGPR1 = K1/K3.
//   B (4x16, KxN): symmetric with lane = N.
//   C/D (16x16):   c[r] holds M = r + 8*(lane>=16), N = lane%16.
//
// All scattered/streamed addresses use 32-bit element offsets against SGPR
// bases (num_nodes*HIDDEN = 12.8M elems, E*HIDDEN = 256M elems < 2^31), so
// the compiler emits SADDR-form vmem ops with no 64-bit VALU address math.
// ---------------------------------------------------------------------------
__global__ __launch_bounds__(256) void rbf_msg_scatter_wmma(
    const float* __restrict__ x,    // [E, HIDDEN]
    const float* __restrict__ rbf,  // [E, NRAD]
    const float* __restrict__ W,    // [HIDDEN, NRAD]  (lin_rbf weight [out,in])
    const int*   __restrict__ idx,  // [E]
    float*       __restrict__ out,  // [num_nodes, HIDDEN]
    int numTiles)                   // E / 16
{
  const int lane = (int)(threadIdx.x & 31u);
  const int wave = (int)(threadIdx.x >> 5u);
  const int half = lane >> 4;   // 0: lanes 0-15, 1: lanes 16-31
  const int lm   = lane & 15;

  const int wavesPerBlock = (int)(blockDim.x >> 5u);
  const int totalWaves    = (int)gridDim.x * wavesPerBlock;
  const int firstTile     = (int)blockIdx.x * wavesPerBlock + wave;

  // ---- Preload B = W^T as 8 N-tiles, K padded 6 -> 8 (rows 6,7 zero) ------
  v2f b0[NTILES], b1[NTILES];
#pragma unroll
  for (int t = 0; t < NTILES; ++t) {
    const float* wrow = W + (unsigned)(t * 16 + lm) * NRAD;
    b0[t].x = wrow[2 * half];                              // K = 0 / 2
    b0[t].y = wrow[2 * half + 1];                          // K = 1 / 3
    b1[t].x = half ? 0.0f : wrow[4];                       // K = 4 / 6(pad)
    b1[t].y = half ? 0.0f : wrow[5];                       // K = 5 / 7(pad)
  }

  for (int tile = firstTile; tile < numTiles; tile += totalWaves) {
    const int eb = tile * 16;

    // ---- A tile: rbf rows eb..eb+15, K padded 6 -> 8 ----------------------
    const float* rrow = rbf + (size_t)(unsigned)(eb + lm) * NRAD;
    v2f a0, a1;
    a0.x = rrow[2 * half];
    a0.y = rrow[2 * half + 1];
    a1.x = half ? 0.0f : rrow[4];
    a1.y = half ? 0.0f : rrow[5];

    // ---- 32-bit destination element offsets for this lane's 8 acc rows ----
    unsigned dstOff[8];
#pragma unroll
    for (int r = 0; r < 8; ++r)
      dstOff[r] = (unsigned)idx[eb + r + 8 * half] * (unsigned)HIDDEN;

    // wave-uniform tile base for x -> SGPR base, 32-bit lane offsets
    const float* __restrict__ xtile = x + (size_t)(unsigned)eb * HIDDEN;

    // ---- 8 N-tiles: 2 WMMAs each, then multiply by x and scatter ----------
#pragma unroll
    for (int t = 0; t < NTILES; ++t) {
      v8f c = {};
      c = __builtin_amdgcn_wmma_f32_16x16x4_f32(false, a0, false, b0[t],
                                                (short)0, c, false, false);
      c = __builtin_amdgcn_wmma_f32_16x16x4_f32(false, a1, false, b1[t],
                                                (short)0, c, false, false);
      const unsigned h = (unsigned)(t * 16 + lm);

      // batch the 8 NT loads so they are all in flight before the first use
      float xv[8];
#pragma unroll
      for (int r = 0; r < 8; ++r)
        xv[r] = __builtin_nontemporal_load(
            &xtile[(unsigned)(r + 8 * half) * HIDDEN + h]);

#pragma unroll
      for (int r = 0; r < 8; ++r)
        atomicAdd(&out[dstOff[r] + h], c[r] * xv[r]);  // non-returning atomic
    }
  }
}

// ---------------------------------------------------------------------------
// Scalar tail for E % 16 != 0 (not hit for E = 2,000,000; kept for generality)
// ---------------------------------------------------------------------------
__global__ void rbf_msg_scatter_tail(
    const float* __restrict__ x, const float* __restrict__ rbf,
    const float* __restrict__ W, const int* __restrict__ idx,
    float* __restrict__ out, long long ebase, int nEdges)
{
  const int tid   = blockIdx.x * blockDim.x + threadIdx.x;
  const int e_off = tid / HIDDEN;
  const int h     = tid % HIDDEN;
  if (e_off >= nEdges) return;
  const long long e = ebase + e_off;
  float s = 0.0f;
#pragma unroll
  for (int r = 0; r < NRAD; ++r)
    s += rbf[e * NRAD + r] * W[(size_t)h * NRAD + r];
  atomicAdd(&out[(long long)idx[e] * HIDDEN + h], s * x[e * HIDDEN + h]);
}

extern "C" void kernel_launch(void* const* d_in, const int* in_sizes, int n_in,
                              void* d_out, int out_size, void* d_ws, size_t ws_size,
                              hipStream_t stream) {
  const float* x   = (const float*)d_in[0];
  const float* rbf = (const float*)d_in[1];
  const float* W   = (const float*)d_in[2];
  const int*   idx = (const int*)d_in[3];
  float*       out = (float*)d_out;

  const long long E = in_sizes[3];   // edge count from index array

  // 1) zero the (poisoned) output
  zero_out_kernel<<<1024, 256, 0, stream>>>(out, out_size);

  // 2) WMMA message + scatter over 16-edge tiles (grid-stride waves)
  const long long numTiles = E / 16;
  if (numTiles > 0) {
    int blocks = 4096;                                   // 32768 persistent waves
    if (numTiles < (long long)blocks * 8)
      blocks = (int)((numTiles + 7) / 8);
    rbf_msg_scatter_wmma<<<blocks, 256, 0, stream>>>(x, rbf, W, idx, out,
                                                     (int)numTiles);
  }

  // 3) tail edges (E % 16), scalar path
  const long long rem = E % 16;
  if (rem > 0) {
    const long long work = rem * HIDDEN;
    const int tblocks = (int)((work + 255) / 256);
    rbf_msg_scatter_tail<<<tblocks, 256, 0, stream>>>(x, rbf, W, idx, out,
                                                      E - rem, (int)rem);
  }
}